// CorrNL_25726854103557
// MI455X (gfx1250) — compile-verified
//
#include <hip/hip_runtime.h>
#include <math.h>

// ---------------------------------------------------------------------------
// Shapes (fixed by the reference): B=64, C=512, H1=16, H2=32, HW=1024,
// POOL=8 -> NC=64 corr channels, INTER=32, RED=16, HW/4=256.
// ---------------------------------------------------------------------------
#define BATCH 64
#define CIN   512
#define H1    16
#define H2    32
#define HW    1024   // 32*32
#define HW4   256    // 16*16
#define POOLN 8
#define NC    64
#define INTER 32
#define RED   16
#define FSCALE (1.0f / 16.0f)
#define BN_EPS 1e-5f

typedef __attribute__((ext_vector_type(2))) float v2f;
typedef __attribute__((ext_vector_type(8))) float v8f;

// fp32 WMMA: D(16x16,f32) = A(16x4,f32) * B(4x16,f32) + C
__device__ __forceinline__ v8f wmma4(v2f a, v2f b, v8f c) {
    return __builtin_amdgcn_wmma_f32_16x16x4_f32(
        /*neg_a=*/false, a, /*neg_b=*/false, b,
        /*c_mod=*/(short)0, c, /*reuse_a=*/false, /*reuse_b=*/false);
}

// A-fragment (row-major A, lda): lane L<16 holds M=m0+L, K={k0,k0+1};
// lanes 16..31 hold M=m0+(L-16), K={k0+2,k0+3}.
__device__ __forceinline__ v2f ldA(const float* __restrict__ A, int m0, int k0,
                                   int lda, int ln, int hi) {
    const float* p = A + (size_t)(m0 + ln) * lda + k0 + 2 * hi;
    v2f a; a[0] = p[0]; a[1] = p[1];
    return a;
}

// B-fragment (row-major B, ldb): VGPR v holds row K=k0+v (lanes 0-15) /
// K=k0+v+2 (lanes 16-31), column N=n0+(lane&15).
__device__ __forceinline__ v2f ldB(const float* __restrict__ B, int k0, int n0,
                                   int ldb, int ln, int hi) {
    const float* p = B + (size_t)(k0 + 2 * hi) * ldb + n0 + ln;
    v2f b; b[0] = p[0]; b[1] = p[ldb];
    return b;
}

__device__ __forceinline__ float tri_int(float u) {
    u = fminf(1.0f, fmaxf(-1.0f, u));
    return (u < 0.0f) ? 0.5f * (u + 1.0f) * (u + 1.0f)
                      : 0.5f + u - 0.5f * u * u;
}

// ---------------------------------------------------------------------------
// K1: per-box separable PrRoI weights wy[b][8][16], wx[b][8][16], area[b]
// ---------------------------------------------------------------------------
__global__ __launch_bounds__(256) void k_boxw(const float* __restrict__ bb,
                                              float* __restrict__ wy,
                                              float* __restrict__ wx,
                                              float* __restrict__ area) {
    int b = blockIdx.x;
    float bx = bb[b * 4 + 0], by = bb[b * 4 + 1];
    float bw_ = bb[b * 4 + 2], bh_ = bb[b * 4 + 3];
    float x1 = bx * FSCALE, y1 = by * FSCALE;
    float x2 = (bx + bw_) * FSCALE, y2 = (by + bh_) * FSCALE;
    float binw = (x2 - x1) * (1.0f / POOLN);
    float binh = (y2 - y1) * (1.0f / POOLN);
    int t = threadIdx.x;
    if (t < 128) {
        int p = t >> 4, h = t & 15;
        float ys = y1 + p * binh, ye = ys + binh;
        wy[b * 128 + t] = tri_int(ye - (float)h) - tri_int(ys - (float)h);
    } else {
        int t2 = t - 128;
        int p = t2 >> 4, w = t2 & 15;
        float xs = x1 + p * binw, xe = xs + binw;
        wx[b * 128 + t2] = tri_int(xe - (float)w) - tri_int(xs - (float)w);
    }
    if (t == 0) area[b] = binw * binh;
}

// ---------------------------------------------------------------------------
// K2: PrRoIPool -> kT[b][k=p*8+q][c]  (A-matrix of the corr GEMM, 64x512)
// block = 256 threads = 4 groups of 64; group handles one channel.
// ---------------------------------------------------------------------------
__global__ __launch_bounds__(256) void k_prroi(const float* __restrict__ feat1,
                                               const float* __restrict__ wy,
                                               const float* __restrict__ wx,
                                               const float* __restrict__ area,
                                               float* __restrict__ kT) {
    int b = blockIdx.y;
    int c0 = blockIdx.x * 4;
    __shared__ float swy[128], swx[128], sf[4][256];
    int t = threadIdx.x;
    if (t < 128) swy[t] = wy[b * 128 + t];
    else         swx[t - 128] = wx[b * 128 + (t - 128)];
    int g = t >> 6, tl = t & 63;
    const float* F = feat1 + ((size_t)b * CIN + c0 + g) * 256;
#pragma unroll
    for (int j = 0; j < 4; ++j) sf[g][tl + 64 * j] = F[tl + 64 * j];
    __syncthreads();
    int p = tl >> 3, q = tl & 7;
    float acc = 0.0f;
#pragma unroll
    for (int h = 0; h < 16; ++h) {
        float row = 0.0f;
#pragma unroll
        for (int w = 0; w < 16; ++w) row += swx[q * 16 + w] * sf[g][h * 16 + w];
        acc += swy[p * 16 + h] * row;
    }
    kT[((size_t)b * (POOLN * POOLN) + (p * POOLN + q)) * CIN + c0 + g] =
        acc / area[b];
}

// ---------------------------------------------------------------------------
// K3: corr[b][64][1024] = kT(64x512) x feat2(512x1024)   [WMMA fp32]
// grid (8, B), block 256 (8 waves x 16 cols = 128 cols per block)
// ---------------------------------------------------------------------------
__global__ __launch_bounds__(256) void k_corr(const float* __restrict__ kT,
                                              const float* __restrict__ feat2,
                                              float* __restrict__ corr) {
    int b = blockIdx.y;
    int lane = threadIdx.x & 31, ln = lane & 15, hi = lane >> 4;
    int n0 = blockIdx.x * 128 + (threadIdx.x >> 5) * 16;
    const float* A = kT + (size_t)b * NC * CIN;
    const float* Bm = feat2 + (size_t)b * CIN * HW;
    v8f acc[4] = {};
    for (int k0 = 0; k0 < CIN; k0 += 4) {
        v2f bf = ldB(Bm, k0, n0, HW, ln, hi);
#pragma unroll
        for (int mt = 0; mt < 4; ++mt) {
            v2f af = ldA(A, mt * 16, k0, CIN, ln, hi);
            acc[mt] = wmma4(af, bf, acc[mt]);
        }
    }
    float* Cp = corr + (size_t)b * NC * HW;
#pragma unroll
    for (int mt = 0; mt < 4; ++mt)
#pragma unroll
        for (int r = 0; r < 8; ++r)
            Cp[(size_t)(mt * 16 + r + 8 * hi) * HW + n0 + ln] = acc[mt][r];
}

// ---------------------------------------------------------------------------
// K4: SE attention e[b][64] = sigmoid(relu(mean(corr) W1^T + b1) W2^T + b2)
// ---------------------------------------------------------------------------
__global__ __launch_bounds__(256) void k_se(const float* __restrict__ corr,
                                            const float* __restrict__ w1,
                                            const float* __restrict__ b1,
                                            const float* __restrict__ w2,
                                            const float* __restrict__ b2,
                                            float* __restrict__ e) {
    int b = blockIdx.x, t = threadIdx.x;
    __shared__ float ss[NC];
    __shared__ float hh[RED];
    if (t < NC) ss[t] = 0.0f;
    __syncthreads();
    const float* cb = corr + (size_t)b * NC * HW;
    for (int k = 0; k < NC; ++k) {
        float s4 = 0.0f;
#pragma unroll
        for (int it = 0; it < 4; ++it) s4 += cb[(size_t)k * HW + it * 256 + t];
        atomicAdd(&ss[k], s4);
    }
    __syncthreads();
    if (t < RED) {
        float h = b1[t];
        for (int k = 0; k < NC; ++k)
            h += (ss[k] * (1.0f / (float)HW)) * w1[t * NC + k];
        hh[t] = fmaxf(h, 0.0f);
    }
    __syncthreads();
    if (t < NC) {
        float o = b2[t];
#pragma unroll
        for (int r = 0; r < RED; ++r) o += hh[r] * w2[t * RED + r];
        e[b * NC + t] = 1.0f / (1.0f + __expf(-o));
    }
}

// ---------------------------------------------------------------------------
// K5: theta/phi/g 1x1 convs over x = corr*e (e fused into B-frag loads)
// A = weight (32x64), B = x (64x1024). grid (8, B), block 256.
// ---------------------------------------------------------------------------
__global__ __launch_bounds__(256) void k_tpg(
    const float* __restrict__ corr, const float* __restrict__ e,
    const float* __restrict__ tw, const float* __restrict__ tb,
    const float* __restrict__ pw, const float* __restrict__ pb,
    const float* __restrict__ gw, const float* __restrict__ gb,
    float* __restrict__ theta, float* __restrict__ phi,
    float* __restrict__ gout) {
    int b = blockIdx.y;
    __shared__ float se_[NC];
    if (threadIdx.x < NC) se_[threadIdx.x] = e[b * NC + threadIdx.x];
    __syncthreads();
    int lane = threadIdx.x & 31, ln = lane & 15, hi = lane >> 4;
    int n0 = blockIdx.x * 128 + (threadIdx.x >> 5) * 16;
    const float* X = corr + (size_t)b * NC * HW;
    v8f at[2] = {}, ap[2] = {}, ag[2] = {};
    for (int k0 = 0; k0 < NC; k0 += 4) {
        int kx = k0 + 2 * hi;
        v2f bf;
        bf[0] = X[(size_t)kx * HW + n0 + ln] * se_[kx];
        bf[1] = X[(size_t)(kx + 1) * HW + n0 + ln] * se_[kx + 1];
#pragma unroll
        for (int mt = 0; mt < 2; ++mt) {
            v2f a;
            a = ldA(tw, mt * 16, k0, NC, ln, hi); at[mt] = wmma4(a, bf, at[mt]);
            a = ldA(pw, mt * 16, k0, NC, ln, hi); ap[mt] = wmma4(a, bf, ap[mt]);
            a = ldA(gw, mt * 16, k0, NC, ln, hi); ag[mt] = wmma4(a, bf, ag[mt]);
        }
    }
    float* T = theta + (size_t)b * INTER * HW;
    float* P = phi + (size_t)b * INTER * HW;
    float* G = gout + (size_t)b * INTER * HW;
#pragma unroll
    for (int mt = 0; mt < 2; ++mt)
#pragma unroll
        for (int r = 0; r < 8; ++r) {
            int m = mt * 16 + r + 8 * hi;
            size_t o = (size_t)m * HW + n0 + ln;
            T[o] = at[mt][r] + tb[m];
            P[o] = ap[mt][r] + pb[m];
            G[o] = ag[mt][r] + gb[m];
        }
}

// ---------------------------------------------------------------------------
// K6: 2x2 maxpool. phi(32x32x32)->phi_p[b][i][m], g->g_t[b][m][i] (transposed)
// ---------------------------------------------------------------------------
__global__ __launch_bounds__(256) void k_pool(const float* __restrict__ phi,
                                              const float* __restrict__ gin,
                                              float* __restrict__ phi_p,
                                              float* __restrict__ g_t) {
    int idx = blockIdx.x * 256 + threadIdx.x;  // BATCH*INTER*HW4
    if (idx >= BATCH * INTER * HW4) return;
    int b = idx / (INTER * HW4);
    int rem = idx - b * (INTER * HW4);
    int i = rem >> 8, m = rem & 255;
    int r2 = m >> 4, c2 = m & 15;
    size_t base = ((size_t)b * INTER + i) * HW + (2 * r2) * H2 + 2 * c2;
    float pv = fmaxf(fmaxf(phi[base], phi[base + 1]),
                     fmaxf(phi[base + H2], phi[base + H2 + 1]));
    float gv = fmaxf(fmaxf(gin[base], gin[base + 1]),
                     fmaxf(gin[base + H2], gin[base + H2 + 1]));
    phi_p[((size_t)b * INTER + i) * HW4 + m] = pv;
    g_t[((size_t)b * HW4 + m) * INTER + i] = gv;
}

// ---------------------------------------------------------------------------
// K7: f[b][n][m] = sum_i theta[b][i][n] * phi_p[b][i][m]
// M=1024(n) N=256(m) K=32(i). grid (8,16,B), block 256 (8 waves stack in M).
// ---------------------------------------------------------------------------
__global__ __launch_bounds__(256) void k_f(const float* __restrict__ theta,
                                           const float* __restrict__ phi_p,
                                           float* __restrict__ f) {
    int b = blockIdx.z;
    int lane = threadIdx.x & 31, ln = lane & 15, hi = lane >> 4;
    int nb = blockIdx.x * 128 + (threadIdx.x >> 5) * 16;
    int m0 = blockIdx.y * 16;
    const float* T = theta + (size_t)b * INTER * HW;
    const float* P = phi_p + (size_t)b * INTER * HW4;
    v8f acc = {};
    for (int k0 = 0; k0 < INTER; k0 += 4) {
        int kx = k0 + 2 * hi;
        v2f af;  // A[n][i] = theta[i][n] (theta is row-major INTERxHW)
        af[0] = T[(size_t)kx * HW + nb + ln];
        af[1] = T[(size_t)(kx + 1) * HW + nb + ln];
        v2f bf = ldB(P, k0, m0, HW4, ln, hi);
        acc = wmma4(af, bf, acc);
    }
    float* F = f + (size_t)b * HW * HW4;
#pragma unroll
    for (int r = 0; r < 8; ++r)
        F[(size_t)(nb + r + 8 * hi) * HW4 + m0 + ln] = acc[r];
}

// ---------------------------------------------------------------------------
// K8: in-place row softmax over 256, one wave per row (wave32 shfl reduce)
// ---------------------------------------------------------------------------
__global__ __launch_bounds__(256) void k_softmax(float* __restrict__ f) {
    int lane = threadIdx.x & 31;
    size_t row = (size_t)blockIdx.x * 8 + (threadIdx.x >> 5);
    float* p = f + row * HW4;
    float v[8];
    float mx = -3.0e38f;
#pragma unroll
    for (int j = 0; j < 8; ++j) { v[j] = p[j * 32 + lane]; mx = fmaxf(mx, v[j]); }
#pragma unroll
    for (int o = 16; o >= 1; o >>= 1) mx = fmaxf(mx, __shfl_xor(mx, o, 32));
    float s = 0.0f;
#pragma unroll
    for (int j = 0; j < 8; ++j) { v[j] = __expf(v[j] - mx); s += v[j]; }
#pragma unroll
    for (int o = 16; o >= 1; o >>= 1) s += __shfl_xor(s, o, 32);
    float inv = 1.0f / s;
#pragma unroll
    for (int j = 0; j < 8; ++j) p[j * 32 + lane] = v[j] * inv;
}

// ---------------------------------------------------------------------------
// K9: y[b][n][i] = sum_m f[b][n][m] * g_t[b][m][i]   M=1024 N=32 K=256
// grid (8, B), block 256; wave -> 16 rows, 2 N-tiles.
// ---------------------------------------------------------------------------
__global__ __launch_bounds__(256) void k_y(const float* __restrict__ f,
                                           const float* __restrict__ g_t,
                                           float* __restrict__ y) {
    int b = blockIdx.y;
    int lane = threadIdx.x & 31, ln = lane & 15, hi = lane >> 4;
    int nb = blockIdx.x * 128 + (threadIdx.x >> 5) * 16;
    const float* F = f + (size_t)b * HW * HW4;
    const float* G = g_t + (size_t)b * HW4 * INTER;
    v8f acc[2] = {};
    for (int k0 = 0; k0 < HW4; k0 += 4) {
        v2f af = ldA(F, nb, k0, HW4, ln, hi);
#pragma unroll
        for (int nt = 0; nt < 2; ++nt) {
            v2f bf = ldB(G, k0, nt * 16, INTER, ln, hi);
            acc[nt] = wmma4(af, bf, acc[nt]);
        }
    }
    float* Y = y + (size_t)b * HW * INTER;
#pragma unroll
    for (int nt = 0; nt < 2; ++nt)
#pragma unroll
        for (int r = 0; r < 8; ++r)
            Y[(size_t)(nb + r + 8 * hi) * INTER + nt * 16 + ln] = acc[nt][r];
}

// ---------------------------------------------------------------------------
// K10: out[b][o][n] = BN(W y^T + Wb)[o][n] + corr[b][o][n]*e[b][o]
// M=64(o) N=1024(n) K=32(i). grid (8, B), block 256.
// ---------------------------------------------------------------------------
__global__ __launch_bounds__(256) void k_out(
    const float* __restrict__ Ww, const float* __restrict__ Wb,
    const float* __restrict__ gamma, const float* __restrict__ beta,
    const float* __restrict__ mean, const float* __restrict__ var,
    const float* __restrict__ y, const float* __restrict__ corr,
    const float* __restrict__ e, float* __restrict__ out) {
    int b = blockIdx.y;
    int lane = threadIdx.x & 31, ln = lane & 15, hi = lane >> 4;
    int n0 = blockIdx.x * 128 + (threadIdx.x >> 5) * 16;
    const float* Y = y + (size_t)b * HW * INTER;
    v8f acc[4] = {};
    for (int k0 = 0; k0 < INTER; k0 += 4) {
        int kx = k0 + 2 * hi;
        v2f bf;  // B[i][n] = y[n][i]
        bf[0] = Y[(size_t)(n0 + ln) * INTER + kx];
        bf[1] = Y[(size_t)(n0 + ln) * INTER + kx + 1];
#pragma unroll
        for (int mt = 0; mt < 4; ++mt) {
            v2f af = ldA(Ww, mt * 16, k0, INTER, ln, hi);
            acc[mt] = wmma4(af, bf, acc[mt]);
        }
    }
    const float* C = corr + (size_t)b * NC * HW;
    const float* E = e + (size_t)b * NC;
    float* O = out + (size_t)b * NC * HW;
#pragma unroll
    for (int mt = 0; mt < 4; ++mt)
#pragma unroll
        for (int r = 0; r < 8; ++r) {
            int o = mt * 16 + r + 8 * hi;
            int n = n0 + ln;
            float val = acc[mt][r] + Wb[o];
            float inv = gamma[o] * rsqrtf(var[o] + BN_EPS);
            val = (val - mean[o]) * inv + beta[o];
            val += C[(size_t)o * HW + n] * E[o];
            O[(size_t)o * HW + n] = val;
        }
}

// ---------------------------------------------------------------------------
// Host launch
// ---------------------------------------------------------------------------
extern "C" void kernel_launch(void* const* d_in, const int* in_sizes, int n_in,
                              void* d_out, int out_size, void* d_ws,
                              size_t ws_size, hipStream_t stream) {
    (void)in_sizes; (void)n_in; (void)out_size; (void)ws_size;
    const float* feat1 = (const float*)d_in[0];
    const float* feat2 = (const float*)d_in[1];
    const float* bb1   = (const float*)d_in[2];
    const float* se_w1 = (const float*)d_in[3];
    const float* se_b1 = (const float*)d_in[4];
    const float* se_w2 = (const float*)d_in[5];
    const float* se_b2 = (const float*)d_in[6];
    const float* tw    = (const float*)d_in[7];
    const float* tb    = (const float*)d_in[8];
    const float* pw    = (const float*)d_in[9];
    const float* pb    = (const float*)d_in[10];
    const float* gw    = (const float*)d_in[11];
    const float* gb    = (const float*)d_in[12];
    const float* nlW   = (const float*)d_in[13];
    const float* nlWb  = (const float*)d_in[14];
    const float* bng   = (const float*)d_in[15];
    const float* bnb   = (const float*)d_in[16];
    const float* bnm   = (const float*)d_in[17];
    const float* bnv   = (const float*)d_in[18];
    float* out = (float*)d_out;
    float* ws = (float*)d_ws;

    // persistent workspace layout (floats)
    size_t o = 0;
    float* wy    = ws + o; o += (size_t)BATCH * 128;
    float* wx    = ws + o; o += (size_t)BATCH * 128;
    float* area  = ws + o; o += 64;
    float* e_    = ws + o; o += (size_t)BATCH * NC;
    float* corr  = ws + o; o += (size_t)BATCH * NC * HW;        // 16 MB
    float* theta = ws + o; o += (size_t)BATCH * INTER * HW;     // 8 MB
    float* phip  = ws + o; o += (size_t)BATCH * INTER * HW4;    // 2 MB
    float* gt    = ws + o; o += (size_t)BATCH * HW4 * INTER;    // 2 MB
    float* y_    = ws + o; o += (size_t)BATCH * HW * INTER;     // 8 MB
    // shared region, reused in phases: kT -> (phi,g) -> f   (64 MB)
    float* kT  = ws + o;
    float* phi = ws + o;
    float* g_  = ws + o + (size_t)BATCH * INTER * HW;
    float* f_  = ws + o;

    k_boxw<<<BATCH, 256, 0, stream>>>(bb1, wy, wx, area);
    k_prroi<<<dim3(CIN / 4, BATCH), 256, 0, stream>>>(feat1, wy, wx, area, kT);
    k_corr<<<dim3(HW / 128, BATCH), 256, 0, stream>>>(kT, feat2, corr);
    k_se<<<BATCH, 256, 0, stream>>>(corr, se_w1, se_b1, se_w2, se_b2, e_);
    k_tpg<<<dim3(HW / 128, BATCH), 256, 0, stream>>>(corr, e_, tw, tb, pw, pb,
                                                     gw, gb, theta, phi, g_);
    k_pool<<<(BATCH * INTER * HW4 + 255) / 256, 256, 0, stream>>>(phi, g_,
                                                                  phip, gt);
    k_f<<<dim3(HW / 128, HW4 / 16, BATCH), 256, 0, stream>>>(theta, phip, f_);
    k_softmax<<<(BATCH * HW) / 8, 256, 0, stream>>>(f_);
    k_y<<<dim3(HW / 128, BATCH), 256, 0, stream>>>(f_, gt, y_);
    k_out<<<dim3(HW / 128, BATCH), 256, 0, stream>>>(nlW, nlWb, bng, bnb, bnm,
                                                     bnv, y_, corr, e_, out);
}